// RotTetrisModel_88656714925194
// MI455X (gfx1250) — compile-verified
//
#include <hip/hip_runtime.h>
#include <math.h>

// ---------------- problem constants (from reference) ----------------
#define NN   100000   // nodes
#define NE   1600000  // edges
#define NG   1000     // graphs
#define REP  16
#define RD   16       // r_dim (bessel)
#define NSC  16       // n_scalars
#define NCL  6        // n_classes
#define LEAK 0.01f
#define PI_F 3.14159265358979323846f
#define SQRT2_F 1.41421356237f

typedef float v2f __attribute__((ext_vector_type(2)));
typedef float v8f __attribute__((ext_vector_type(8)));

__device__ __forceinline__ float leaky(float x) { return x > 0.f ? x : LEAK * x; }

// ---------------- Tensor Data Mover: 1-D tile Global -> LDS ----------------
#if defined(__has_builtin)
#  if __has_builtin(__builtin_amdgcn_tensor_load_to_lds) && __has_builtin(__builtin_amdgcn_s_wait_tensorcnt)
#    define HAVE_TDM 1
#  endif
#endif
#ifndef HAVE_TDM
#  define HAVE_TDM 0
#endif

#if HAVE_TDM
typedef unsigned int u32x4 __attribute__((ext_vector_type(4)));
typedef int          i32x8 __attribute__((ext_vector_type(8)));
typedef int          i32x4 __attribute__((ext_vector_type(4)));

// Copy nfloats (must be < 65536) contiguous floats from global to LDS via TDM.
// D# layout per CDNA5 ISA ch.8 (group0: count|lds_addr|global_addr|type=2,
// group1: data_size=4B, tensor_dim0=tile_dim0=n, tile_dim1=1, dim0_stride=n).
// This toolchain exposes the 6-arg builtin (clang-23 / therock-10.0 lane):
//   (u32x4 g0, i32x8 g1, i32x4 g2, i32x4 g3, i32x8 extra, i32 cpol)
__device__ __forceinline__ void tdm_copy_1d(const float* gsrc, float* lds_dst, int nfloats) {
    unsigned long long ga = (unsigned long long)(uintptr_t)gsrc;
    unsigned int la = (unsigned int)(uintptr_t)lds_dst;   // low 32 bits of generic = LDS byte addr
    unsigned int n = (unsigned int)nfloats;
    u32x4 g0;
    g0[0] = 1u;                                               // count=1 (valid), no gather
    g0[1] = la;                                               // lds_addr (bits 63:32)
    g0[2] = (unsigned int)(ga & 0xFFFFFFFFu);                 // global_addr[31:0]
    g0[3] = (unsigned int)((ga >> 32) & 0x1FFFFFFu) | 0x80000000u; // global_addr[56:32], type=2
    i32x8 g1;
    g1[0] = (int)(2u << 16);                                  // workgroup_mask=0, data_size=4B
    g1[1] = (int)((n & 0xFFFFu) << 16);                       // tensor_dim0[15:0]
    g1[2] = (int)(((n >> 16) & 0xFFFFu) | (1u << 16));        // tensor_dim0[31:16], tensor_dim1=1
    g1[3] = (int)((n & 0xFFFFu) << 16);                       // tile_dim0 = n
    g1[4] = 1;                                                // tile_dim1 = 1, tile_dim2 = 0
    g1[5] = (int)n;                                           // tensor_dim0_stride[31:0]
    g1[6] = 0;
    g1[7] = 0;
    i32x4 z4 = {0, 0, 0, 0};
    i32x8 z8 = {0, 0, 0, 0, 0, 0, 0, 0};
    __builtin_amdgcn_tensor_load_to_lds(g0, g1, z4, z4, z8, 0);
}
#endif

// ---------------- utility: zero a float region ----------------
__global__ void k_zero(float* p, int n) {
    int i = blockIdx.x * blockDim.x + threadIdx.x;
    if (i < n) p[i] = 0.f;
}

// ---------------- per-graph mean accumulation ----------------
__global__ void k_graph_accum(const float* __restrict__ pos, const int* __restrict__ batch,
                              float* gsum, float* gcnt) {
    int n = blockIdx.x * blockDim.x + threadIdx.x;
    if (n >= NN) return;
    int b = batch[n];
    atomicAdd(&gsum[2 * b + 0], pos[2 * n + 0]);
    atomicAdd(&gsum[2 * b + 1], pos[2 * n + 1]);
    atomicAdd(&gcnt[b], 1.f);
}

// ---------------- center positions, per-node rotation frame ----------------
__global__ void k_center(const float* __restrict__ pos, const int* __restrict__ batch,
                         const float* __restrict__ gsum, const float* __restrict__ gcnt,
                         float* posg, float* thc, float* ths) {
    int n = blockIdx.x * blockDim.x + threadIdx.x;
    if (n >= NN) return;
    int b = batch[n];
    float c = fmaxf(gcnt[b], 1.f);
    float px = pos[2 * n + 0] - gsum[2 * b + 0] / c;
    float py = pos[2 * n + 1] - gsum[2 * b + 1] / c;
    posg[2 * n + 0] = px;
    posg[2 * n + 1] = py;
    float r = sqrtf(px * px + py * py);
    if (r > 0.f) { thc[n] = px / r; ths[n] = py / r; }
    else         { thc[n] = 1.f;    ths[n] = 0.f;   }
}

// ---------------- per-edge geometry + embedding scatter ----------------
// e_rot[e,j] = R_e * w_j  =>  edge sum factors into per-node (sum cos, sum sin)
__global__ void k_edge_embed(const float* __restrict__ posg, const int* __restrict__ ei,
                             float* ec, float* es, float* edist,
                             float* deg, float* dsum, float* crc, float* crs) {
    int e = blockIdx.x * blockDim.x + threadIdx.x;
    if (e >= NE) return;
    int r0 = ei[e], c0 = ei[NE + e];
    float vx = posg[2 * r0 + 0] - posg[2 * c0 + 0];
    float vy = posg[2 * r0 + 1] - posg[2 * c0 + 1];
    float d = sqrtf(vx * vx + vy * vy);
    float rc, rs;
    if (d > 0.f) { rc = vx / d; rs = vy / d; } else { rc = 1.f; rs = 0.f; }
    ec[e] = rc; es[e] = rs; edist[e] = d;
    atomicAdd(&deg[r0], 1.f);
    atomicAdd(&dsum[r0], d);
    atomicAdd(&crc[r0], rc);
    atomicAdd(&crs[r0], rs);
}

// ---------------- node feature init ----------------
__global__ void k_node_init(const float* __restrict__ Wemb, const float* __restrict__ wse,
                            const float* __restrict__ bse,
                            const float* __restrict__ deg, const float* __restrict__ dsum,
                            const float* __restrict__ crc, const float* __restrict__ crs,
                            float* x_rot, float* x_s) {
    int n = blockIdx.x * blockDim.x + threadIdx.x;
    if (n >= NN) return;
    float dv = fmaxf(deg[n], 1.f);
    float Sc = crc[n], Ss = crs[n];
#pragma unroll
    for (int j = 0; j < REP; j++) {
        float w0 = Wemb[2 * j + 0], w1 = Wemb[2 * j + 1];
        x_rot[n * 32 + 2 * j + 0] = (Sc * w0 - Ss * w1) / dv;
        x_rot[n * 32 + 2 * j + 1] = (Ss * w0 + Sc * w1) / dv;
    }
    float dm = dsum[n] / dv;
#pragma unroll
    for (int k = 0; k < NSC; k++) x_s[n * 16 + k] = leaky(dm * wse[k] + bse[k]);
}

// ---------------- per-layer edge kernel ----------------
__global__ void k_edge_layer(const int* __restrict__ ei,
                             const float* __restrict__ ec, const float* __restrict__ es,
                             const float* __restrict__ edist,
                             const float* __restrict__ x_rot, const float* __restrict__ x_s,
                             const float* __restrict__ Wg_i, const float* __restrict__ bg_i,
                             const float* __restrict__ Wmix_i,
                             const float* __restrict__ Ws1_i, const float* __restrict__ bs1_i,
                             float* agg, float* sagg) {
    __shared__ float s_wg[256], s_wmix[256], s_ws1[512], s_bg[16], s_bs1[16];
    int tid = threadIdx.x;
#if HAVE_TDM
    if (tid < 32) {   // wave 0 issues the tensor DMA for all three weight blocks
        tdm_copy_1d(Wg_i,   s_wg,   256);
        tdm_copy_1d(Wmix_i, s_wmix, 256);
        tdm_copy_1d(Ws1_i,  s_ws1,  512);
        __builtin_amdgcn_s_wait_tensorcnt(0);
    }
#else
    if (tid < 256) { s_wg[tid] = Wg_i[tid]; s_wmix[tid] = Wmix_i[tid]; }
    s_ws1[tid] = Ws1_i[tid];
    s_ws1[tid + 256] = Ws1_i[tid + 256];
#endif
    if (tid < 16) { s_bg[tid] = bg_i[tid]; s_bs1[tid] = bs1_i[tid]; }
    __syncthreads();

    int e = blockIdx.x * blockDim.x + tid;
    if (e >= NE) return;
    int r0 = ei[e], c0 = ei[NE + e];
    float d = edist[e], rc = ec[e], rs = es[e];
    float inv_ds = 1.f / (d + 1e-8f);

    // Bessel basis via Chebyshev recurrence: sin((k+1)x) = 2cos(x)sin(kx) - sin((k-1)x)
    float demb[RD];
    {
        float s1, c1;
        sincosf(PI_F * d, &s1, &c1);
        float twoc = 2.f * c1;
        float skm1 = 0.f, sk = s1;
        demb[0] = SQRT2_F * sk * inv_ds;
#pragma unroll
        for (int k = 1; k < RD; k++) {
            float sn = twoc * sk - skm1;
            skm1 = sk; sk = sn;
            demb[k] = SQRT2_F * sk * inv_ds;
        }
    }

    // rotate-in: local = R^T * x_rot[col]   (vectorized 128B gather)
    float lx[REP], ly[REP];
    {
        const float4* xr4 = (const float4*)(x_rot + (size_t)c0 * 32);
#pragma unroll
        for (int q = 0; q < 8; q++) {
            float4 v = xr4[q];
            int j = 2 * q;
            lx[j]     =  rc * v.x + rs * v.y;
            ly[j]     = -rs * v.x + rc * v.y;
            lx[j + 1] =  rc * v.z + rs * v.w;
            ly[j + 1] = -rs * v.z + rc * v.w;
        }
    }

    // radial gains g = dist_emb @ Wg + bg
    float gv[REP];
#pragma unroll
    for (int r = 0; r < REP; r++) {
        float a = s_bg[r];
#pragma unroll
        for (int k = 0; k < RD; k++) a += demb[k] * s_wg[k * 16 + r];
        gv[r] = a;
    }

    // mix across reps, scale, rotate-out, scatter
#pragma unroll
    for (int j = 0; j < REP; j++) {
        float mx = 0.f, my = 0.f;
#pragma unroll
        for (int k = 0; k < REP; k++) {
            float wv = s_wmix[j * 16 + k];
            mx += wv * lx[k];
            my += wv * ly[k];
        }
        mx *= gv[j]; my *= gv[j];
        float ox = rc * mx - rs * my;
        float oy = rs * mx + rc * my;
        atomicAdd(&agg[r0 * 32 + 2 * j + 0], ox);
        atomicAdd(&agg[r0 * 32 + 2 * j + 1], oy);
    }

    // scalar message: leaky([x_s[col], demb] @ Ws1 + bs1)
    float xs[NSC];
    {
        const float4* xs4 = (const float4*)(x_s + (size_t)c0 * 16);
#pragma unroll
        for (int q = 0; q < 4; q++) {
            float4 v = xs4[q];
            xs[4 * q + 0] = v.x; xs[4 * q + 1] = v.y;
            xs[4 * q + 2] = v.z; xs[4 * q + 3] = v.w;
        }
    }
#pragma unroll
    for (int k = 0; k < NSC; k++) {
        float a = s_bs1[k];
#pragma unroll
        for (int j = 0; j < 16; j++) a += xs[j] * s_ws1[j * 16 + k];
#pragma unroll
        for (int j = 0; j < 16; j++) a += demb[j] * s_ws1[(16 + j) * 16 + k];
        atomicAdd(&sagg[r0 * 16 + k], leaky(a));
    }
}

// ---------------- per-layer node update ----------------
__global__ void k_node_update(float* x_rot, float* x_s,
                              const float* __restrict__ agg, const float* __restrict__ sagg,
                              const float* __restrict__ deg,
                              const float* __restrict__ Wgate_i, const float* __restrict__ bgate_i) {
    __shared__ float s_wgate[256], s_bgate[16];
    int tid = threadIdx.x;
    if (tid < 256) s_wgate[tid] = Wgate_i[tid];
    if (tid < 16) s_bgate[tid] = bgate_i[tid];
    __syncthreads();

    int n = blockIdx.x * blockDim.x + tid;
    if (n >= NN) return;
    float xs[NSC];
#pragma unroll
    for (int k = 0; k < NSC; k++) xs[k] = x_s[n * 16 + k];
    float dv = fmaxf(deg[n], 1.f);
#pragma unroll
    for (int r = 0; r < REP; r++) {
        float a = s_bgate[r];
#pragma unroll
        for (int k = 0; k < NSC; k++) a += xs[k] * s_wgate[k * 16 + r];
        float gate = 1.f / (1.f + expf(-a));
        x_rot[n * 32 + 2 * r + 0] += agg[n * 32 + 2 * r + 0] / dv * gate;
        x_rot[n * 32 + 2 * r + 1] += agg[n * 32 + 2 * r + 1] / dv * gate;
    }
#pragma unroll
    for (int k = 0; k < NSC; k++) x_s[n * 16 + k] = xs[k] + sagg[n * 16 + k] / dv;
}

// ---------------- output MLP: WMMA GEMM [16 nodes x 48] @ [48 x 144] ----------------
// 2 waves per block, 1 tile of 16 nodes per wave. N = 100000 = 6250 * 16 exactly.
#define NTILES 6250
__global__ void __launch_bounds__(64) k_mlp(const float* __restrict__ x_rot,
                                            const float* __restrict__ x_s,
                                            const float* __restrict__ thc,
                                            const float* __restrict__ ths,
                                            const int* __restrict__ batch,
                                            const float* __restrict__ W1, const float* __restrict__ b1,
                                            const float* __restrict__ W2, const float* __restrict__ b2,
                                            float* out) {
    __shared__ float s_w1[48 * 144];
    __shared__ float s_b1[144];
    __shared__ float s_xc[2][16 * 48];    // per-wave x_cat tile
    __shared__ float s_h[2][16 * 144];    // per-wave hidden tile

    int tid = threadIdx.x;
#if HAVE_TDM
    if (tid < 32) {   // wave 0: tensor DMA the 27KB W1 block into LDS
        tdm_copy_1d(W1, s_w1, 48 * 144);
        __builtin_amdgcn_s_wait_tensorcnt(0);
    }
#else
    for (int i = tid; i < 48 * 144; i += 64) s_w1[i] = W1[i];
#endif
    for (int i = tid; i < 144; i += 64) s_b1[i] = b1[i];
    __syncthreads();

    int wave = tid >> 5;
    int lane = tid & 31;
    int tile = blockIdx.x * 2 + wave;
    if (tile >= NTILES) return;
    int m0 = tile * 16;
    float* xc = s_xc[wave];
    float* hh = s_h[wave];

    // stage x_cat = [x_s(16) | (R_theta * x_rot)(32)] for 16 nodes
    for (int idx = lane; idx < 16 * 48; idx += 32) {
        int ml = idx / 48, k = idx % 48;
        int node = m0 + ml;
        float v;
        if (k < 16) {
            v = x_s[node * 16 + k];
        } else {
            int kk = k - 16, j = kk >> 1, comp = kk & 1;
            float x0 = x_rot[node * 32 + 2 * j + 0];
            float x1 = x_rot[node * 32 + 2 * j + 1];
            float c = thc[node], s = ths[node];
            v = comp ? (s * x0 + c * x1) : (c * x0 - s * x1);
        }
        xc[idx] = v;
    }

    int lrow = lane & 15;       // A: M row / B,D: N col
    int lhi  = lane >> 4;       // K half selector (0 -> K+0/1, 1 -> K+2/3)

    // GEMM1: h = leaky(x_cat @ W1 + b1), 9 col-tiles, K = 48 as 12 chained k4 WMMAs
    for (int nt = 0; nt < 9; nt++) {
        int ncol = nt * 16 + lrow;
        float bv = s_b1[ncol];
        v8f acc = {bv, bv, bv, bv, bv, bv, bv, bv};
#pragma unroll
        for (int kc = 0; kc < 12; kc++) {
            int kb = kc * 4 + 2 * lhi;
            v2f a, b;
            a.x = xc[lrow * 48 + kb];
            a.y = xc[lrow * 48 + kb + 1];
            b.x = s_w1[kb * 144 + ncol];
            b.y = s_w1[(kb + 1) * 144 + ncol];
            acc = __builtin_amdgcn_wmma_f32_16x16x4_f32(
                false, a, false, b, (short)0, acc, false, false);
        }
#pragma unroll
        for (int v = 0; v < 8; v++) {
            float h = acc[v];
            h = h > 0.f ? h : LEAK * h;
            int mrow = v + 8 * lhi;
            hh[mrow * 144 + ncol] = h;
        }
    }

    // GEMM2 (tiny N=6): u = h @ W2 + b2, then atomic scatter into per-graph output
#pragma unroll
    for (int t = 0; t < 3; t++) {
        int o = lane + 32 * t;          // 96 = 16 nodes * 6 classes
        int ml = o / 6, cls = o % 6;
        float a = b2[cls];
        for (int k = 0; k < 144; k++) a += hh[ml * 144 + k] * W2[k * 6 + cls];
        int node = m0 + ml;
        int g = batch[node];
        atomicAdd(&out[g * 6 + cls], a);
    }
}

// ---------------- host-side launch ----------------
static inline int cdiv(int a, int b) { return (a + b - 1) / b; }

extern "C" void kernel_launch(void* const* d_in, const int* in_sizes, int n_in,
                              void* d_out, int out_size, void* d_ws, size_t ws_size,
                              hipStream_t stream) {
    const float* pos   = (const float*)d_in[0];
    const float* Wemb  = (const float*)d_in[1];   // [16,1,2]
    const float* wse   = (const float*)d_in[2];   // [16]
    const float* bse   = (const float*)d_in[3];   // [16]
    const float* Wg    = (const float*)d_in[4];   // [3,16,16]
    const float* bg    = (const float*)d_in[5];   // [3,16]
    const float* Wmix  = (const float*)d_in[6];   // [3,16,16]
    const float* Wgate = (const float*)d_in[7];   // [3,16,16]
    const float* bgate = (const float*)d_in[8];   // [3,16]
    const float* Ws1   = (const float*)d_in[9];   // [3,32,16]
    const float* bs1   = (const float*)d_in[10];  // [3,16]
    const float* W1    = (const float*)d_in[11];  // [48,144]
    const float* b1    = (const float*)d_in[12];  // [144]
    const float* W2    = (const float*)d_in[13];  // [144,6]
    const float* b2    = (const float*)d_in[14];  // [6]
    const int*   ei    = (const int*)d_in[15];    // [2,E]
    const int*   batch = (const int*)d_in[16];    // [N]
    float* out = (float*)d_out;

    // workspace layout (floats)
    float* w = (float*)d_ws;
    float* posg  = w;                 // 2N
    float* thc   = w + 2 * (size_t)NN;
    float* ths   = w + 3 * (size_t)NN;
    float* deg   = w + 4 * (size_t)NN;   // deg,dsum,crc,crs contiguous (4N)
    float* dsum  = w + 5 * (size_t)NN;
    float* crc   = w + 6 * (size_t)NN;
    float* crs   = w + 7 * (size_t)NN;
    float* x_rot = w + 8 * (size_t)NN;   // 32N
    float* x_s   = w + 40 * (size_t)NN;  // 16N
    float* agg   = w + 56 * (size_t)NN;  // 32N (agg,sagg contiguous 48N)
    float* sagg  = w + 88 * (size_t)NN;  // 16N
    float* ec    = w + 104 * (size_t)NN;
    float* es    = ec + (size_t)NE;
    float* edist = ec + 2 * (size_t)NE;
    float* gsum  = ec + 3 * (size_t)NE;  // 2G (gsum,gcnt contiguous 3G)
    float* gcnt  = gsum + 2 * (size_t)NG;

    // 1) graph means
    k_zero<<<cdiv(3 * NG, 256), 256, 0, stream>>>(gsum, 3 * NG);
    k_zero<<<cdiv(4 * NN, 256), 256, 0, stream>>>(deg, 4 * NN);
    k_graph_accum<<<cdiv(NN, 256), 256, 0, stream>>>(pos, batch, gsum, gcnt);
    k_center<<<cdiv(NN, 256), 256, 0, stream>>>(pos, batch, gsum, gcnt, posg, thc, ths);

    // 2) edge geometry + embedding scatter
    k_edge_embed<<<cdiv(NE, 256), 256, 0, stream>>>(posg, ei, ec, es, edist,
                                                    deg, dsum, crc, crs);
    k_node_init<<<cdiv(NN, 256), 256, 0, stream>>>(Wemb, wse, bse, deg, dsum, crc, crs,
                                                   x_rot, x_s);

    // 3) equivariant layers
    for (int i = 0; i < 3; i++) {
        k_zero<<<cdiv(48 * NN, 256), 256, 0, stream>>>(agg, 48 * NN);
        k_edge_layer<<<cdiv(NE, 256), 256, 0, stream>>>(
            ei, ec, es, edist, x_rot, x_s,
            Wg + i * 256, bg + i * 16, Wmix + i * 256,
            Ws1 + i * 512, bs1 + i * 16, agg, sagg);
        k_node_update<<<cdiv(NN, 256), 256, 0, stream>>>(
            x_rot, x_s, agg, sagg, deg, Wgate + i * 256, bgate + i * 16);
    }

    // 4) output MLP (WMMA) + per-graph reduction
    k_zero<<<cdiv(NG * NCL, 256), 256, 0, stream>>>(out, NG * NCL);
    k_mlp<<<cdiv(NTILES, 2), 64, 0, stream>>>(x_rot, x_s, thc, ths, batch,
                                              W1, b1, W2, b2, out);
}